// PC_74234214744264
// MI455X (gfx1250) — compile-verified
//
#include <hip/hip_runtime.h>

typedef float v2f __attribute__((ext_vector_type(2)));
typedef float v4f __attribute__((ext_vector_type(4)));
typedef float v8f __attribute__((ext_vector_type(8)));

#define NB   2048
#define TOPG 0.07f
#define BOTG 0.07f
#define N_ITER 25
#define BLS  160   // Bl k-pair row stride (floats); BLS % 64 == 32 -> conflict-free halves

// ---------------- WMMA helpers (fp32 16x16x4, CDNA5) ----------------
__device__ __forceinline__ v8f wmma4(v2f a, v2f b, v8f c) {
  // (neg_a, A, neg_b, B, c_mod, C, reuse_a, reuse_b)
  return __builtin_amdgcn_wmma_f32_16x16x4_f32(false, a, false, b, (short)0, c, false, false);
}

// Stage a 64(K) x 64(N) B tile from global (row stride = `stride` floats) into
// K-pair interleaved LDS: Bl[kp*BLS + n*2 + {0,1}] = B[2kp + {0,1}][n].
__device__ __forceinline__ void stage_b(float* Bl, const float* src, int stride, int tid) {
  for (int idx = tid; idx < 512; idx += 256) {
    int kp = idx >> 4;             // k-pair 0..31
    int n4 = (idx & 15) * 4;       // n group of 4
    v4f r0 = *(const v4f*)(src + (size_t)(2 * kp) * stride + n4);
    v4f r1 = *(const v4f*)(src + (size_t)(2 * kp + 1) * stride + n4);
    float* d = &Bl[kp * BLS + n4 * 2];
    v2f t;
    t.x = r0.x; t.y = r1.x; *(v2f*)(d + 0) = t;
    t.x = r0.y; t.y = r1.y; *(v2f*)(d + 2) = t;
    t.x = r0.z; t.y = r1.z; *(v2f*)(d + 4) = t;
    t.x = r0.w; t.y = r1.w; *(v2f*)(d + 6) = t;
  }
}

// One K-block (64) of WMMAs: 16 k-steps x 4 N-tiles, per-wave 16x64 output strip.
// Al stride = 68 floats (conflict-free, 16B-aligned rows); B operands are single
// ds_load_b64 from the pair-interleaved Bl.
__device__ __forceinline__ void mma_block(const float* Al, const float* Bl, v8f acc[4],
                                          int mr, int half, int lq) {
#pragma unroll
  for (int kk = 0; kk < 64; kk += 4) {
    v2f a = *(const v2f*)&Al[(mr + lq) * 68 + kk + 2 * half];
    const float* bp = &Bl[((kk >> 1) + half) * BLS + lq * 2];
#pragma unroll
    for (int nt = 0; nt < 4; ++nt) {
      v2f b = *(const v2f*)(bp + nt * 32);
      acc[nt] = wmma4(a, b, acc[nt]);
    }
  }
}

// ---------------- weight reordering (once per launch) ----------------
// W2f[tap][c64][o128], W2b[tap][o128][c64] from W2 (OIHW 128x64x5x5)
// W3f[tap][c128][o256], W3b[tap][o256][c128] from W3 (OIHW 256x128x3x3)
__global__ void k_prep_w(const float* __restrict__ W2, const float* __restrict__ W3,
                         float* __restrict__ W2f, float* __restrict__ W2b,
                         float* __restrict__ W3f, float* __restrict__ W3b) {
  int i = blockIdx.x * 256 + threadIdx.x;
  if (i < 204800) {
    { int o = i & 127, c = (i >> 7) & 63, tap = i >> 13;  // 64*128 = 2^13
      W2f[i] = W2[(o * 64 + c) * 25 + tap]; }
    { int c = i & 63, o = (i >> 6) & 127, tap = i >> 13;
      W2b[i] = W2[(o * 64 + c) * 25 + tap]; }
  }
  if (i < 294912) {
    { int o = i & 255, c = (i >> 8) & 127, tap = i >> 15;  // 128*256 = 2^15
      W3f[i] = W3[(o * 128 + c) * 9 + tap]; }
    { int c = i & 127, o = (i >> 7) & 255, tap = i >> 15;
      W3b[i] = W3[(o * 128 + c) * 9 + tap]; }
  }
}

// ---------------- conv1 (runs once; tiny FLOPs, direct VALU) ----------------
// x: NCHW [B,3,32,32]; h1,t1: NHWC [B,14,14,64]
__global__ void __launch_bounds__(256) k_conv1(const float* __restrict__ x,
                                               const float* __restrict__ W1,
                                               const float* __restrict__ b1,
                                               float* __restrict__ h1, float* __restrict__ t1) {
  __shared__ float xs[3072];
  const int b = blockIdx.x, tid = threadIdx.x;
  for (int idx = tid; idx < 3072; idx += 256) xs[idx] = x[(size_t)b * 3072 + idx];
  __syncthreads();
  for (int idx = tid; idx < 12544; idx += 256) {
    int c = idx & 63, pos = idx >> 6;
    int oy = pos / 14, ox = pos % 14;
    float s = b1[c];
#pragma unroll
    for (int ci = 0; ci < 3; ++ci)
#pragma unroll
      for (int ky = 0; ky < 5; ++ky)
#pragma unroll
        for (int kx = 0; kx < 5; ++kx)
          s += xs[ci * 1024 + (2 * oy + ky) * 32 + (2 * ox + kx)] *
               W1[((c * 3 + ci) * 5 + ky) * 5 + kx];
    s = s > 0.f ? s : 0.f;
    size_t a = (size_t)b * 12544 + (size_t)pos * 64 + c;
    h1[a] = s; t1[a] = s;
  }
}

// ---------------- conv2 forward: p2 = relu(conv(t1,W2)+b2) ----------------
// M = B*25 (rows = (b,oy,ox)), N = 128, K = 25 taps x 64c. grid (400, 2).
__global__ void __launch_bounds__(256) k_conv2_fwd(const float* __restrict__ t1,
                                                   const float* __restrict__ W2f,
                                                   const float* __restrict__ b2,
                                                   float* __restrict__ p2,
                                                   float* __restrict__ t2also) {
  __shared__ float Al[128 * 68];
  __shared__ float Bl[32 * BLS];
  const int tid = threadIdx.x, lane = tid & 31, wv = tid >> 5;
  const int half = lane >> 4, lq = lane & 15;
  const int r0 = blockIdx.x * 128, n0 = blockIdx.y * 64, mr = wv * 16;
  v8f acc[4];
  for (int nt = 0; nt < 4; ++nt) for (int e = 0; e < 8; ++e) acc[nt][e] = 0.f;

  for (int tap = 0; tap < 25; ++tap) {
    const int ky = tap / 5, kx = tap % 5;
    __syncthreads();
    for (int idx = tid; idx < 128 * 16; idx += 256) {
      int row = idx >> 4, q = idx & 15;
      int r = r0 + row, b = r / 25, s = r % 25, oy = s / 5, ox = s % 5;
      v4f v = *(const v4f*)(t1 + ((size_t)(b * 14 + 2 * oy + ky) * 14 + 2 * ox + kx) * 64 + q * 4);
      *(v4f*)&Al[row * 68 + q * 4] = v;
    }
    stage_b(Bl, W2f + (size_t)tap * 8192 + n0, 128, tid);
    __syncthreads();
    mma_block(Al, Bl, acc, mr, half, lq);
  }
  for (int nt = 0; nt < 4; ++nt) {
    int n = n0 + nt * 16 + lq;
    float bias = b2[n];
    for (int e = 0; e < 8; ++e) {
      int r = r0 + mr + e + 8 * half;
      float v = acc[nt][e] + bias;
      v = v > 0.f ? v : 0.f;
      p2[(size_t)r * 128 + n] = v;
      if (t2also) t2also[(size_t)r * 128 + n] = v;
    }
  }
}

// ---------------- conv2 VJP + t1 update ----------------
// Rows in parity-class order: cls=(py,px), rc = b*49 + iy*7 + ix, y=2iy+py, x=2ix+px.
// g1 = sum over valid taps of e2m @ W2b[tap]; t1 = .93 t1 + .07 h1 + .07 g1. grid (3136).
__global__ void __launch_bounds__(256) k_conv2_bwd(const float* __restrict__ e2m,
                                                   const float* __restrict__ W2b,
                                                   const float* __restrict__ h1,
                                                   float* __restrict__ t1) {
  __shared__ float Al[128 * 68];
  __shared__ float Bl[32 * BLS];
  const int tid = threadIdx.x, lane = tid & 31, wv = tid >> 5;
  const int half = lane >> 4, lq = lane & 15;
  const int r0 = blockIdx.x * 128;
  const int cls = r0 / (NB * 49), py = cls >> 1, px = cls & 1;
  const int rc0 = r0 % (NB * 49), mr = wv * 16;
  v8f acc[4];
  for (int nt = 0; nt < 4; ++nt) for (int e = 0; e < 8; ++e) acc[nt][e] = 0.f;

  for (int i = 0; i < 3; ++i) {
    int ky = py + 2 * i; if (ky > 4) continue;
    for (int j = 0; j < 3; ++j) {
      int kx = px + 2 * j; if (kx > 4) continue;
      int tap = ky * 5 + kx;
      for (int ks = 0; ks < 2; ++ks) {  // K = 128 (o) in two 64-blocks
        __syncthreads();
        for (int idx = tid; idx < 128 * 16; idx += 256) {
          int row = idx >> 4, q = idx & 15;
          int rc = rc0 + row, b = rc / 49, s = rc % 49;
          int oy = s / 7 - i, ox = s % 7 - j;
          v4f v; v.x = v.y = v.z = v.w = 0.f;
          if (oy >= 0 && oy < 5 && ox >= 0 && ox < 5)
            v = *(const v4f*)(e2m + ((size_t)(b * 5 + oy) * 5 + ox) * 128 + ks * 64 + q * 4);
          *(v4f*)&Al[row * 68 + q * 4] = v;
        }
        stage_b(Bl, W2b + (size_t)tap * 8192 + ks * 64 * 64, 64, tid);
        __syncthreads();
        mma_block(Al, Bl, acc, mr, half, lq);
      }
    }
  }
  for (int nt = 0; nt < 4; ++nt) {
    int c = nt * 16 + lq;
    for (int e = 0; e < 8; ++e) {
      int rc = rc0 + mr + e + 8 * half;
      int b = rc / 49, s = rc % 49;
      int y = (s / 7) * 2 + py, x = (s % 7) * 2 + px;
      size_t a = ((size_t)(b * 14 + y) * 14 + x) * 64 + c;
      t1[a] = (1.f - TOPG) * t1[a] + TOPG * h1[a] + BOTG * acc[nt][e];
    }
  }
}

// ---------------- conv3 forward: p3 = relu(conv(t2,W3)+b3), NCHW-flat out ----------
// M = B*4, N = 256, K = 9 taps x 128c. grid (64, 4).
__global__ void __launch_bounds__(256) k_conv3_fwd(const float* __restrict__ t2,
                                                   const float* __restrict__ W3f,
                                                   const float* __restrict__ b3,
                                                   float* __restrict__ p3,
                                                   float* __restrict__ t3also) {
  __shared__ float Al[128 * 68];
  __shared__ float Bl[32 * BLS];
  const int tid = threadIdx.x, lane = tid & 31, wv = tid >> 5;
  const int half = lane >> 4, lq = lane & 15;
  const int r0 = blockIdx.x * 128, n0 = blockIdx.y * 64, mr = wv * 16;
  v8f acc[4];
  for (int nt = 0; nt < 4; ++nt) for (int e = 0; e < 8; ++e) acc[nt][e] = 0.f;

  for (int tap = 0; tap < 9; ++tap) {
    const int ky = tap / 3, kx = tap % 3;
    for (int ks = 0; ks < 2; ++ks) {  // K = 128 (c) in two 64-blocks
      __syncthreads();
      for (int idx = tid; idx < 128 * 16; idx += 256) {
        int row = idx >> 4, q = idx & 15;
        int r = r0 + row, b = r >> 2, s = r & 3, oy = s >> 1, ox = s & 1;
        v4f v = *(const v4f*)(t2 + ((size_t)(b * 5 + 2 * oy + ky) * 5 + 2 * ox + kx) * 128 +
                              ks * 64 + q * 4);
        *(v4f*)&Al[row * 68 + q * 4] = v;
      }
      stage_b(Bl, W3f + (size_t)tap * 32768 + ks * 64 * 256 + n0, 256, tid);
      __syncthreads();
      mma_block(Al, Bl, acc, mr, half, lq);
    }
  }
  for (int nt = 0; nt < 4; ++nt) {
    int o = n0 + nt * 16 + lq;
    float bias = b3[o];
    for (int e = 0; e < 8; ++e) {
      int r = r0 + mr + e + 8 * half, b = r >> 2, s = r & 3;
      float v = acc[nt][e] + bias;
      v = v > 0.f ? v : 0.f;
      size_t a = (size_t)b * 1024 + o * 4 + s;  // NCHW flatten c*4 + y*2 + x
      p3[a] = v;
      if (t3also) t3also[a] = v;
    }
  }
}

// ---------------- conv3 VJP + t2 update ----------------
// Class rows padded to 3x3 (iy,ix in 0..2); invalid y,x>4 rows computed, not stored.
// t2 = .93 t2 + .07 p2 + .07 g2. grid (576, 2).
__global__ void __launch_bounds__(256) k_conv3_bwd(const float* __restrict__ e3m,
                                                   const float* __restrict__ W3b,
                                                   const float* __restrict__ p2,
                                                   float* __restrict__ t2) {
  __shared__ float Al[128 * 68];
  __shared__ float Bl[32 * BLS];
  const int tid = threadIdx.x, lane = tid & 31, wv = tid >> 5;
  const int half = lane >> 4, lq = lane & 15;
  const int r0 = blockIdx.x * 128, n0 = blockIdx.y * 64;
  const int cls = r0 / (NB * 9), py = cls >> 1, px = cls & 1;
  const int rc0 = r0 % (NB * 9), mr = wv * 16;
  v8f acc[4];
  for (int nt = 0; nt < 4; ++nt) for (int e = 0; e < 8; ++e) acc[nt][e] = 0.f;

  for (int i = 0; i < 2; ++i) {
    int ky = py + 2 * i; if (ky > 2) continue;
    for (int j = 0; j < 2; ++j) {
      int kx = px + 2 * j; if (kx > 2) continue;
      int tap = ky * 3 + kx;
      for (int ks = 0; ks < 4; ++ks) {  // K = 256 (o) in four 64-blocks
        __syncthreads();
        for (int idx = tid; idx < 128 * 16; idx += 256) {
          int row = idx >> 4, q = idx & 15;
          int rc = rc0 + row, b = rc / 9, s = rc % 9;
          int oy = s / 3 - i, ox = s % 3 - j;
          v4f v; v.x = v.y = v.z = v.w = 0.f;
          if (oy >= 0 && oy < 2 && ox >= 0 && ox < 2)
            v = *(const v4f*)(e3m + ((size_t)(b * 2 + oy) * 2 + ox) * 256 + ks * 64 + q * 4);
          *(v4f*)&Al[row * 68 + q * 4] = v;
        }
        stage_b(Bl, W3b + (size_t)tap * 32768 + ks * 64 * 128 + n0, 128, tid);
        __syncthreads();
        mma_block(Al, Bl, acc, mr, half, lq);
      }
    }
  }
  for (int nt = 0; nt < 4; ++nt) {
    int c = n0 + nt * 16 + lq;
    for (int e = 0; e < 8; ++e) {
      int rc = rc0 + mr + e + 8 * half;
      int b = rc / 9, s = rc % 9;
      int y = (s / 3) * 2 + py, x = (s % 3) * 2 + px;
      if (y < 5 && x < 5) {
        size_t a = ((size_t)(b * 5 + y) * 5 + x) * 128 + c;
        t2[a] = (1.f - TOPG) * t2[a] + TOPG * p2[a] + BOTG * acc[nt][e];
      }
    }
  }
}

// ---------------- e2m = (t2 - p2) * (p2 > 0) ----------------
__global__ void k_e2m(const float* __restrict__ t2, const float* __restrict__ p2,
                      float* __restrict__ e2m) {
  size_t i = ((size_t)blockIdx.x * 256 + threadIdx.x) * 4;
  v4f t = *(const v4f*)(t2 + i);
  v4f p = *(const v4f*)(p2 + i);
  v4f r;
  r.x = p.x > 0.f ? t.x - p.x : 0.f;
  r.y = p.y > 0.f ? t.y - p.y : 0.f;
  r.z = p.z > 0.f ? t.z - p.z : 0.f;
  r.w = p.w > 0.f ? t.w - p.w : 0.f;
  *(v4f*)(e2m + i) = r;
}

// ---------------- f4 path: p4 = f4(t3); e4; g3 = e4@W4; e3m; t3 update -------------
__global__ void __launch_bounds__(256) k_f4_step(const float* __restrict__ gt,
                                                 const float* __restrict__ W4,
                                                 const float* __restrict__ b4,
                                                 const float* __restrict__ p3,
                                                 float* __restrict__ t3,
                                                 float* __restrict__ e3m) {
  __shared__ float red[10 * 256];
  __shared__ float p4s[10];
  const int b = blockIdx.x, tid = threadIdx.x;
  float t3r[4], part[10];
#pragma unroll
  for (int j = 0; j < 10; ++j) part[j] = 0.f;
#pragma unroll
  for (int q = 0; q < 4; ++q) {
    int k = q * 256 + tid;
    float tv = t3[(size_t)b * 1024 + k];
    t3r[q] = tv;
#pragma unroll
    for (int j = 0; j < 10; ++j) part[j] += tv * W4[j * 1024 + k];
  }
  for (int j = 0; j < 10; ++j) red[j * 256 + tid] = part[j];
  __syncthreads();
  for (int s = 128; s > 0; s >>= 1) {
    if (tid < s)
      for (int j = 0; j < 10; ++j) red[j * 256 + tid] += red[j * 256 + tid + s];
    __syncthreads();
  }
  if (tid < 10) p4s[tid] = red[tid * 256] + b4[tid];
  __syncthreads();
  float pl[10], mx = -3.402823e38f;
  for (int j = 0; j < 10; ++j) { pl[j] = p4s[j]; mx = fmaxf(mx, pl[j]); }
  float sum = 0.f;
  for (int j = 0; j < 10; ++j) { pl[j] = expf(pl[j] - mx); sum += pl[j]; }
  float inv = 1.f / sum;
  float e4[10];
  for (int j = 0; j < 10; ++j) e4[j] = (gt[(size_t)b * 10 + j] - pl[j] * inv) * 0.5f;
#pragma unroll
  for (int q = 0; q < 4; ++q) {
    int k = q * 256 + tid;
    float g3 = 0.f;
#pragma unroll
    for (int j = 0; j < 10; ++j) g3 += e4[j] * W4[j * 1024 + k];
    float pv = p3[(size_t)b * 1024 + k];
    float e3 = t3r[q] - pv;
    int c = k >> 2, s2 = k & 3;
    e3m[((size_t)(b * 2 + (s2 >> 1)) * 2 + (s2 & 1)) * 256 + c] = pv > 0.f ? e3 : 0.f;
    t3[(size_t)b * 1024 + k] = t3r[q] - TOPG * e3 + BOTG * g3;
  }
}

// ---------------- final output: NHWC->NCHW gather into [B, 16768] ----------------
__global__ void k_out(const float* __restrict__ t1, const float* __restrict__ t2,
                      const float* __restrict__ t3, float* __restrict__ out) {
  size_t i = (size_t)blockIdx.x * 256 + threadIdx.x;
  int b = (int)(i / 16768), q = (int)(i % 16768);
  float v;
  if (q < 12544) {
    int c = q / 196, s = q % 196;
    v = t1[(size_t)b * 12544 + (size_t)s * 64 + c];
  } else if (q < 15744) {
    int q2 = q - 12544;
    int c = q2 / 25, s = q2 % 25;
    v = t2[(size_t)b * 3200 + (size_t)s * 128 + c];
  } else {
    v = t3[(size_t)b * 1024 + (q - 15744)];
  }
  out[i] = v;
}

extern "C" void kernel_launch(void* const* d_in, const int* in_sizes, int n_in,
                              void* d_out, int out_size, void* d_ws, size_t ws_size,
                              hipStream_t stream) {
  (void)in_sizes; (void)n_in; (void)out_size; (void)ws_size;
  const float* x  = (const float*)d_in[0];
  const float* gt = (const float*)d_in[1];
  const float* W1 = (const float*)d_in[2];
  const float* b1 = (const float*)d_in[3];
  const float* W2 = (const float*)d_in[4];
  const float* b2 = (const float*)d_in[5];
  const float* W3 = (const float*)d_in[6];
  const float* b3 = (const float*)d_in[7];
  const float* W4 = (const float*)d_in[8];
  const float* b4 = (const float*)d_in[9];
  float* ws  = (float*)d_ws;
  float* h1  = ws;                //  25,690,112  NHWC [B,14,14,64]
  float* t1  = h1  + 25690112;    //  25,690,112
  float* t2  = t1  + 25690112;    //   6,553,600  NHWC [B,5,5,128]
  float* p2  = t2  + 6553600;     //   6,553,600
  float* e2m = p2  + 6553600;     //   6,553,600
  float* t3  = e2m + 6553600;     //   2,097,152  [B,1024] NCHW-flat
  float* p3  = t3  + 2097152;     //   2,097,152
  float* e3m = p3  + 2097152;     //   2,097,152  NHWC [B,2,2,256]
  float* W2f = e3m + 2097152;     //     204,800
  float* W2b = W2f + 204800;      //     204,800
  float* W3f = W2b + 204800;      //     294,912
  float* W3b = W3f + 294912;      //     294,912   (total ~313 MB)
  float* outp = (float*)d_out;

  k_prep_w<<<1152, 256, 0, stream>>>(W2, W3, W2f, W2b, W3f, W3b);
  k_conv1<<<NB, 256, 0, stream>>>(x, W1, b1, h1, t1);
  k_conv2_fwd<<<dim3(400, 2), 256, 0, stream>>>(t1, W2f, b2, p2, t2);  // t2 = h2
  k_conv3_fwd<<<dim3(64, 4), 256, 0, stream>>>(t2, W3f, b3, p3, t3);   // t3 = h3

  for (int it = 0; it < N_ITER; ++it) {
    k_f4_step<<<NB, 256, 0, stream>>>(gt, W4, b4, p3, t3, e3m);            // e3m, t3+=
    k_e2m<<<6400, 256, 0, stream>>>(t2, p2, e2m);                           // e2m
    k_conv3_bwd<<<dim3(576, 2), 256, 0, stream>>>(e3m, W3b, p2, t2);        // t2+=
    k_conv2_bwd<<<3136, 256, 0, stream>>>(e2m, W2b, h1, t1);                // t1+=
    k_conv2_fwd<<<dim3(400, 2), 256, 0, stream>>>(t1, W2f, b2, p2, nullptr);
    k_conv3_fwd<<<dim3(64, 4), 256, 0, stream>>>(t2, W3f, b3, p3, nullptr);
  }
  k_out<<<134144, 256, 0, stream>>>(t1, t2, t3, outp);
}